// MultiTaskRankingLoss_65377992179983
// MI455X (gfx1250) — compile-verified
//
#include <hip/hip_runtime.h>
#include <math.h>

#define BB 384
#define D_EMB 256
#define NUM_TASKS 4
#define MARGIN_F 1.0f
#define EPS_F 1e-6f

typedef __attribute__((ext_vector_type(2))) float v2f;
typedef __attribute__((ext_vector_type(8))) float v8f;

// ---------------------------------------------------------------------------
// Kernel 0: per-row squared norms and element sums; zero global accumulators.
// ---------------------------------------------------------------------------
__global__ __launch_bounds__(128) void prep_kernel(const float* __restrict__ emb,
                                                   float* __restrict__ norms,
                                                   float* __restrict__ rsums,
                                                   float* __restrict__ sacc,
                                                   unsigned* __restrict__ cacc) {
    int i = blockIdx.x * blockDim.x + threadIdx.x;
    if (i < NUM_TASKS) { sacc[i] = 0.0f; cacc[i] = 0u; }
    if (i < BB) {
        const float* row = emb + i * D_EMB;
        float n = 0.0f, s = 0.0f;
#pragma unroll 4
        for (int k = 0; k < D_EMB; ++k) {
            float x = row[k];
            n = fmaf(x, x, n);
            s += x;
        }
        norms[i] = n;
        rsums[i] = s;
    }
}

// ---------------------------------------------------------------------------
// Kernel 1: dist[i,j] via f32 WMMA Gram matrix.
// One wave (32 lanes) per 16x16 output tile. K accumulated 4 at a time with
// V_WMMA_F32_16X16X4_F32.
//   A (16x4 f32): lane l holds row m=l%16; VGPR0=K=(l<16?0:2), VGPR1=K+1.
//   B (4x16 f32): same pattern with columns n=l%16 (B = X^T rows of j-tile).
//   C/D (16x16 f32): VGPR r -> M=r (lanes 0-15) / M=r+8 (lanes 16-31), N=l%16.
// dist^2 = G_ii + G_jj - 2 G_ij + 2*eps*(S_i - S_j) + D*eps^2
// ---------------------------------------------------------------------------
__global__ __launch_bounds__(32) void dist_kernel(const float* __restrict__ emb,
                                                  const float* __restrict__ norms,
                                                  const float* __restrict__ rsums,
                                                  float* __restrict__ dist) {
    const int i0   = blockIdx.x * 16;
    const int j0   = blockIdx.y * 16;
    const int lane = threadIdx.x;   // 0..31
    const int half = lane >> 4;     // 0 or 1
    const int l16  = lane & 15;

    const float* arow = emb + (i0 + l16) * D_EMB + 2 * half;
    const float* brow = emb + (j0 + l16) * D_EMB + 2 * half;

    v8f acc = {};
#pragma unroll 8
    for (int k0 = 0; k0 < D_EMB; k0 += 4) {
        v2f a = *(const v2f*)(arow + k0);   // 8-byte aligned: k0 even
        v2f b = *(const v2f*)(brow + k0);
        acc = __builtin_amdgcn_wmma_f32_16x16x4_f32(
            /*neg_a=*/false, a, /*neg_b=*/false, b,
            /*c_mod=*/(short)0, acc, /*reuse_a=*/false, /*reuse_b=*/false);
    }

    const int   n  = j0 + l16;
    const float nj = norms[n];
    const float sj = rsums[n];
#pragma unroll
    for (int r = 0; r < 8; ++r) {
        const int m = i0 + r + 8 * half;
        float d2 = norms[m] + nj - 2.0f * acc[r]
                 + 2.0f * EPS_F * (rsums[m] - sj)
                 + (float)D_EMB * EPS_F * EPS_F;
        dist[m * BB + n] = sqrtf(fmaxf(d2, 0.0f));
    }
}

// ---------------------------------------------------------------------------
// Kernel 2: B^3 masked triple accumulation. One block per anchor i.
// LDS-resident: dist row, packed rankings (4 tasks x u8 per int), and the
// per-j bitmask of tasks with r_i > r_j (whole j-row skipped if zero).
// ---------------------------------------------------------------------------
__device__ inline unsigned pack4(const int* __restrict__ r) {
    return ((unsigned)(r[0] & 255))        | ((unsigned)(r[1] & 255) << 8) |
           ((unsigned)(r[2] & 255) << 16)  | ((unsigned)(r[3] & 255) << 24);
}

__global__ __launch_bounds__(256) void triple_kernel(const float* __restrict__ dist,
                                                     const int* __restrict__ rank,
                                                     float* __restrict__ sacc,
                                                     unsigned* __restrict__ cacc) {
    __shared__ float    drow[BB];
    __shared__ unsigned prank[BB];
    __shared__ unsigned mij[BB];
    __shared__ float    ssum[NUM_TASKS];
    __shared__ unsigned scnt[NUM_TASKS];

    const int i   = blockIdx.x;
    const int tid = threadIdx.x;

    const unsigned pi = pack4(rank + i * NUM_TASKS);

    for (int j = tid; j < BB; j += blockDim.x) {
        drow[j] = dist[i * BB + j];
        unsigned p = pack4(rank + j * NUM_TASKS);
        prank[j] = p;
        unsigned m = 0;
#pragma unroll
        for (int t = 0; t < NUM_TASKS; ++t)
            if (((pi >> (8 * t)) & 255u) > ((p >> (8 * t)) & 255u)) m |= (1u << t);
        mij[j] = m;
    }
    if (tid < NUM_TASKS) { ssum[tid] = 0.0f; scnt[tid] = 0u; }
    __syncthreads();

    float    s0 = 0.f, s1 = 0.f, s2 = 0.f, s3 = 0.f;
    unsigned c0 = 0,   c1 = 0,   c2 = 0,   c3 = 0;

    for (int j = tid; j < BB; j += blockDim.x) {
        const unsigned m = mij[j];
        if (!m) continue;                    // no task has r_i > r_j
        const float    dj = drow[j];
        const unsigned pj = prank[j];
        for (int k = 0; k < BB; ++k) {
            const float    tr = fmaxf(dj - drow[k] + MARGIN_F, 0.0f);
            const unsigned pk = prank[k];
            if ((m & 1u) && ((pj        & 255u) > (pk        & 255u))) { s0 += tr; ++c0; }
            if ((m & 2u) && (((pj >> 8 ) & 255u) > ((pk >> 8 ) & 255u))) { s1 += tr; ++c1; }
            if ((m & 4u) && (((pj >> 16) & 255u) > ((pk >> 16) & 255u))) { s2 += tr; ++c2; }
            if ((m & 8u) && (((pj >> 24) & 255u) > ((pk >> 24) & 255u))) { s3 += tr; ++c3; }
        }
    }

    // Block-level reduction via LDS atomics, then one global atomic per task.
    atomicAdd(&ssum[0], s0); atomicAdd(&ssum[1], s1);
    atomicAdd(&ssum[2], s2); atomicAdd(&ssum[3], s3);
    atomicAdd(&scnt[0], c0); atomicAdd(&scnt[1], c1);
    atomicAdd(&scnt[2], c2); atomicAdd(&scnt[3], c3);
    __syncthreads();
    if (tid < NUM_TASKS) {
        atomicAdd(&sacc[tid], ssum[tid]);
        atomicAdd(&cacc[tid], scnt[tid]);
    }
}

// ---------------------------------------------------------------------------
// Kernel 3: finalize scalar loss.
// ---------------------------------------------------------------------------
__global__ void finalize_kernel(const float* __restrict__ sacc,
                                const unsigned* __restrict__ cacc,
                                const float* __restrict__ w,
                                float* __restrict__ out) {
    if (threadIdx.x == 0 && blockIdx.x == 0) {
        float total = 0.0f;
#pragma unroll
        for (int t = 0; t < NUM_TASKS; ++t) {
            float c = (float)cacc[t];
            if (c > 0.0f) total += w[t] * (sacc[t] / fmaxf(c, 1.0f));
        }
        out[0] = total;
    }
}

// ---------------------------------------------------------------------------
extern "C" void kernel_launch(void* const* d_in, const int* in_sizes, int n_in,
                              void* d_out, int out_size, void* d_ws, size_t ws_size,
                              hipStream_t stream) {
    (void)in_sizes; (void)n_in; (void)out_size; (void)ws_size;

    const float* emb  = (const float*)d_in[0];   // (384, 256) f32
    const float* w    = (const float*)d_in[1];   // (4,) f32
    const int*   rank = (const int*)d_in[2];     // (384, 4) i32
    float*       out  = (float*)d_out;           // scalar f32

    // Workspace layout (floats): dist[B*B] | norms[B] | rsums[B] | sacc[4] | cacc[4]
    float*    dist  = (float*)d_ws;
    float*    norms = dist + BB * BB;
    float*    rsums = norms + BB;
    float*    sacc  = rsums + BB;
    unsigned* cacc  = (unsigned*)(sacc + NUM_TASKS);

    prep_kernel<<<dim3((BB + 127) / 128), dim3(128), 0, stream>>>(emb, norms, rsums, sacc, cacc);
    dist_kernel<<<dim3(BB / 16, BB / 16), dim3(32), 0, stream>>>(emb, norms, rsums, dist);
    triple_kernel<<<dim3(BB), dim3(256), 0, stream>>>(dist, rank, sacc, cacc);
    finalize_kernel<<<1, 32, 0, stream>>>(sacc, cacc, w, out);
}